// PermutationClosedLayer_14276471292438
// MI455X (gfx1250) — compile-verified
//
#include <hip/hip_runtime.h>

// PermutationClosedLayer on gfx1250:
//   out[b, p*64+m] = sum_n x[b,n] * W[p,m,n],  W[p,m,n] = (m==n) ? w[p,0] : w[p,1]
// => GEMM (B x 64) * (64 x 512) with the 512x64 weight matrix synthesized from
//    16 scalars in registers. Memory-bound (512 MiB out + 64 MiB in); compute
//    done with chained V_WMMA_F32_16X16X4_F32 to keep full f32 precision.

typedef __attribute__((ext_vector_type(2))) float v2f;
typedef __attribute__((ext_vector_type(8))) float v8f;

__global__ __launch_bounds__(256) void pcl_wmma_f32_kernel(
    const float* __restrict__ x,   // [nrows, 64]
    const float* __restrict__ w,   // [8, 2]
    float* __restrict__ out,       // [nrows, 512]
    int nrows)
{
    const int lane = threadIdx.x & 31;
    const int wave = threadIdx.x >> 5;
    const int tile = blockIdx.x * 8 + wave;          // 16-row tile index
    const long long row0 = (long long)tile * 16;
    if (row0 >= nrows) return;                       // wave-uniform: EXEC stays all-1s

    // Broadcast the 16 weight scalars (uniform address -> scalar loads).
    float w0r[8], w1r[8];
#pragma unroll
    for (int p = 0; p < 8; ++p) {
        w0r[p] = w[2 * p + 0];   // diagonal weight
        w1r[p] = w[2 * p + 1];   // off-diagonal weight
    }

    const int m16 = lane & 15;       // A: row M / B,D: column N / D: col
    const int hi  = lane >> 4;       // lane half selects K pair {0,1} vs {2,3}

    // ---- Load A fragments: 16x64 f32 tile of x, ISA 16x4 A layout ----
    // VGPR0 of frag kk holds K = 4*kk + 2*hi, VGPR1 holds K = 4*kk + 2*hi + 1.
    const float* xrow = x + (row0 + m16) * 64 + 2 * hi;
    v2f a[16];
#pragma unroll
    for (int kk = 0; kk < 16; ++kk)
        a[kk] = *(const v2f*)(xrow + 4 * kk);        // global_load_b64

    // D-tile store base: C/D layout -> VGPR v: rows v (lanes 0-15) / v+8 (16-31)
    float* obase = out + (row0 + 8ll * hi) * 512 + m16;

    // ---- 32 output column tiles: p in [0,8) x q in [0,4), cols p*64+q*16+n ----
#pragma unroll
    for (int p = 0; p < 8; ++p) {
        const float W0 = w0r[p];
        const float W1 = w1r[p];
        for (int q = 0; q < 4; ++q) {
            const int mcol = q * 16 + m16;           // logical W-row of this column
            v8f acc = {};
#pragma unroll
            for (int kk = 0; kk < 16; ++kk) {
                // Synthesize B (K x N = 4x16) fragment: B[k][n] = (mcol==k)?W0:W1
                const int kb = 4 * kk + 2 * hi;
                v2f bfrag;
                bfrag.x = (mcol == kb)     ? W0 : W1;
                bfrag.y = (mcol == kb + 1) ? W0 : W1;
                // D = A(16x4) * B(4x16) + C, chained f32 accumulation over K=64
                acc = __builtin_amdgcn_wmma_f32_16x16x4_f32(
                    false, a[kk], false, bfrag, (short)0, acc, false, false);
            }
            // Stream the 16x16 f32 tile: 8 stores, 2x64B contiguous per store
            float* op = obase + (p * 64 + q * 16);
#pragma unroll
            for (int v = 0; v < 8; ++v)
                op[(long long)v * 512] = acc[v];
        }
    }
}

extern "C" void kernel_launch(void* const* d_in, const int* in_sizes, int n_in,
                              void* d_out, int out_size, void* d_ws, size_t ws_size,
                              hipStream_t stream) {
    const float* x = (const float*)d_in[0];   // [B, 64] f32
    const float* w = (const float*)d_in[1];   // [8, 2] f32
    // d_in[2] (indices = 1 - eye(64)) is never read: encoded analytically.
    float* out = (float*)d_out;               // [B, 512] f32

    const int nrows  = in_sizes[0] / 64;      // B = 262144
    const int tiles  = (nrows + 15) / 16;     // 16-row tiles, one wave each
    const int blocks = (tiles + 7) / 8;       // 8 waves (256 threads) per block

    pcl_wmma_f32_kernel<<<blocks, 256, 0, stream>>>(x, w, out, nrows);
}